// QuantizerEMA_53360673685832
// MI455X (gfx1250) — compile-verified
//
#include <hip/hip_runtime.h>
#include <hip/hip_bf16.h>

// VQ-VAE quantizer forward on gfx1250 using V_WMMA_F32_16X16X4_F32.
// d2(n,k) = ||z_n||^2 - 2 z_n.e_k + ||e_k||^2 ; argmin_k needs only (||e||^2 - 2 z.e).
// Round 2: hoisted B-fragment loads + dual accumulator chains (2 code-tiles/iter)
// to hide LDS latency and keep the matrix pipe fed.

typedef float v2f __attribute__((ext_vector_type(2)));
typedef float v8f __attribute__((ext_vector_type(8)));

#define N_TOTAL   65536     // 16*64*64 latent vectors
#define K_CODES   4096
#define DIM       64
#define CHUNK     128       // codes staged in LDS per iteration
#define CHUNKS    (K_CODES / CHUNK)          // 32
#define TILES     (CHUNK / 16)               // 8 code-tiles per chunk
#define EPAD      68        // padded dword stride per staged code (bank-conflict free)
#define ROWS_PER_BLOCK 128  // 8 waves * 16 rows
#define NBLOCKS   (N_TOTAL / ROWS_PER_BLOCK) // 512
#define OUT_IDX_OFF  (N_TOTAL * DIM)         // 4194304
#define LOSS_OFF     (OUT_IDX_OFF + N_TOTAL) // 4259840

__global__ __launch_bounds__(256) void vq_argmin_kernel(
    const float* __restrict__ z,        // [N_TOTAL, 64]
    const float* __restrict__ emb,      // [4096, 64]
    float* __restrict__ out,            // quantized(BDHW) | indices | loss
    float* __restrict__ wsPartial)      // [NBLOCKS] per-block loss partials
{
    __shared__ float ldsE[CHUNK * EPAD];   // staged codes, padded
    __shared__ float eeLds[CHUNK];         // ||e||^2 per staged code
    __shared__ int   idxArr[8][16];        // argmin per wave-row
    __shared__ float lossArr[8];           // per-wave loss partial

    const int tid  = threadIdx.x;
    const int lane = tid & 31;
    const int w    = tid >> 5;             // wave id 0..7
    const int h    = (lane >> 4) & 1;      // lane half
    const int n16  = lane & 15;

    const int rowBase = blockIdx.x * ROWS_PER_BLOCK + w * 16;
    const int bImg    = rowBase >> 12;     // 4096 rows per image
    const int hw0     = rowBase & 4095;

    // ---- Preload A fragments: z-tile rows (16x64 fp32), 32 regs/lane ----
    // A layout (16x4 fp32): lane holds M=lane%16; VGPR0=K(2h), VGPR1=K(2h+1) per chunk.
    float az[32];
    {
        const float* zrow = z + (size_t)(rowBase + n16) * DIM;
        #pragma unroll
        for (int kk = 0; kk < 16; ++kk) {
            float2 t2 = *(const float2*)&zrow[kk * 4 + 2 * h];
            az[2 * kk]     = t2.x;
            az[2 * kk + 1] = t2.y;
        }
    }
    // ||z_row||^2 : lane holds half the row; other half sits in lane^16.
    float zz;
    {
        float p = 0.f;
        #pragma unroll
        for (int i = 0; i < 32; ++i) p += az[i] * az[i];
        zz = p + __shfl_xor(p, 16, 32);
    }

    // Running min of s = ||e||^2 - 2 z.e  (per C/D slot v: row M = v + 8*h, col N = lane%16)
    float best[8];
    int   bidx[8];
    #pragma unroll
    for (int v = 0; v < 8; ++v) { best[v] = 3.4e38f; bidx[v] = 0; }

    for (int c = 0; c < CHUNKS; ++c) {
        __syncthreads();   // previous chunk fully consumed

        // ---- Stage CHUNK codes into LDS (coalesced b128 loads / ds_store_b128) ----
        {
            const float4* embV = (const float4*)(emb + (size_t)c * CHUNK * DIM);
            #pragma unroll
            for (int it = 0; it < (CHUNK * DIM / 4) / 256; ++it) {   // 8 iters
                int i4   = it * 256 + tid;
                int dw   = i4 * 4;
                int code = dw >> 6;          // /64
                int dimS = dw & 63;
                float4 v4 = embV[i4];
                *(float4*)&ldsE[code * EPAD + dimS] = v4;
            }
        }
        __syncthreads();

        // ---- ||e||^2 for staged codes ----
        if (tid < CHUNK) {
            const float* er = &ldsE[tid * EPAD];
            float s = 0.f;
            #pragma unroll
            for (int d = 0; d < DIM; ++d) s += er[d] * er[d];
            eeLds[tid] = s;
        }
        __syncthreads();

        // ---- 2 code-tiles per iteration: hoisted B loads, dual WMMA chains ----
        for (int t = 0; t < TILES; t += 2) {
            const int base0 = (t * 16 + n16) * EPAD + 2 * h;
            const int base1 = base0 + 16 * EPAD;

            v2f b0[16], b1[16];
            #pragma unroll
            for (int kk = 0; kk < 16; ++kk) {
                b0[kk] = *(const v2f*)&ldsE[base0 + kk * 4];
                b1[kk] = *(const v2f*)&ldsE[base1 + kk * 4];
            }
            const float e20 = eeLds[t * 16 + n16];
            const float e21 = eeLds[t * 16 + 16 + n16];

            v8f acc0 = {};
            v8f acc1 = {};
            #pragma unroll
            for (int kk = 0; kk < 16; ++kk) {
                v2f a; a.x = az[2 * kk]; a.y = az[2 * kk + 1];
                acc0 = __builtin_amdgcn_wmma_f32_16x16x4_f32(
                        false, a, false, b0[kk], (short)0, acc0, false, false);
                acc1 = __builtin_amdgcn_wmma_f32_16x16x4_f32(
                        false, a, false, b1[kk], (short)0, acc1, false, false);
            }

            const int code0 = c * CHUNK + t * 16 + n16;
            const int code1 = code0 + 16;
            #pragma unroll
            for (int v = 0; v < 8; ++v) {
                float s0 = e20 - 2.0f * acc0[v];
                float s1 = e21 - 2.0f * acc1[v];
                if (s0 < best[v]) { best[v] = s0; bidx[v] = code0; }  // ascending scan
                if (s1 < best[v]) { best[v] = s1; bidx[v] = code1; }  // => first min kept
            }
        }
    }

    // ---- Cross-lane argmin within each 16-lane half (fixed M per slot) ----
    #pragma unroll
    for (int v = 0; v < 8; ++v) {
        #pragma unroll
        for (int m = 1; m < 16; m <<= 1) {
            float ob = __shfl_xor(best[v], m, 32);
            int   oi = __shfl_xor(bidx[v], m, 32);
            if (ob < best[v] || (ob == best[v] && oi < bidx[v])) { best[v] = ob; bidx[v] = oi; }
        }
    }

    // Lanes 0..7 own rows 0..7 (slot=lane); lanes 24..31 own rows 8..15 (slot=lane-24).
    int   myIdx  = 0;
    float myBest = 0.f;
    #pragma unroll
    for (int v = 0; v < 8; ++v) {
        bool sel = (lane == v) || (lane == v + 24);
        if (sel) { myIdx = bidx[v]; myBest = best[v]; }
    }
    const bool owner = (lane < 8) || (lane >= 24);
    if (lane < 8)        idxArr[w][lane]      = myIdx;
    else if (lane >= 24) idxArr[w][lane - 16] = myIdx;

    // per-row loss = ||z||^2 + min_s ; wave sum
    float contrib = owner ? (zz + myBest) : 0.f;
    #pragma unroll
    for (int m = 1; m < 32; m <<= 1) contrib += __shfl_xor(contrib, m, 32);
    if (lane == 0) lossArr[w] = contrib;

    __syncthreads();

    if (tid == 0) {
        float s = 0.f;
        #pragma unroll
        for (int i = 0; i < 8; ++i) s += lossArr[i];
        wsPartial[blockIdx.x] = s;
    }

    // ---- Write indices (as float, reference dtype-coerced) ----
    if (lane < 16) out[OUT_IDX_OFF + rowBase + lane] = (float)idxArr[w][lane];

    // ---- Write quantized, transposed to [B, D, H, W]: contiguous 64B per half-wave ----
    const int myCode = idxArr[w][n16];          // code for row r = lane%16
    #pragma unroll
    for (int it = 0; it < 32; ++it) {
        int d = 2 * it + h;
        float val = emb[(size_t)myCode * DIM + d];
        size_t addr = (((size_t)(bImg * DIM + d)) << 12) + (size_t)(hw0 + n16);
        out[addr] = val;
    }
}

__global__ __launch_bounds__(512) void vq_loss_reduce_kernel(
    const float* __restrict__ wsPartial, float* __restrict__ out)
{
    __shared__ float sm[512];
    int t = threadIdx.x;
    sm[t] = wsPartial[t];
    __syncthreads();
    #pragma unroll
    for (int s = 256; s > 0; s >>= 1) {
        if (t < s) sm[t] += sm[t + s];
        __syncthreads();
    }
    if (t == 0) out[LOSS_OFF] = 0.25f * sm[0];
}

extern "C" void kernel_launch(void* const* d_in, const int* in_sizes, int n_in,
                              void* d_out, int out_size, void* d_ws, size_t ws_size,
                              hipStream_t stream) {
    (void)in_sizes; (void)n_in; (void)out_size; (void)ws_size;
    const float* z   = (const float*)d_in[0];
    const float* emb = (const float*)d_in[1];
    float* out = (float*)d_out;
    float* ws  = (float*)d_ws;

    vq_argmin_kernel<<<NBLOCKS, 256, 0, stream>>>(z, emb, out, ws);
    vq_loss_reduce_kernel<<<1, 512, 0, stream>>>(ws, out);
}